// CRFLayer_74981539053998
// MI455X (gfx1250) — compile-verified
//
#include <hip/hip_runtime.h>

#define DIMS   48
#define TMAX   512
#define CHUNK  64                       // timesteps per prefetch chunk
#define BLOCK  64                       // 2 waves (wave32)
#define CHUNK_BYTES (CHUNK * DIMS * 4)  // 12288 B per chunk

#if defined(__has_builtin)
#if __has_builtin(__builtin_amdgcn_tensor_load_to_lds) && __has_builtin(__builtin_amdgcn_s_wait_tensorcnt)
#define USE_TDM 1
#endif
#endif
#ifndef USE_TDM
#define USE_TDM 0
#endif

typedef unsigned int u32x4 __attribute__((ext_vector_type(4)));
typedef int          i32x8 __attribute__((ext_vector_type(8)));
typedef int          i32x4 __attribute__((ext_vector_type(4)));

// ---------------------------------------------------------------------------
// Path A (preferred): Tensor Data Mover. One instruction DMAs the whole
// 48x64-element (12 KB) emission tile into LDS; completion via TENSORcnt.
// 6-arg builtin variant (clang-23 / therock-10.0 headers).
// ---------------------------------------------------------------------------
#if USE_TDM
__device__ __forceinline__ void tdm_load_chunk(const float* row, int c, float* dst) {
    unsigned long long ga = (unsigned long long)(const void*)(row + (size_t)c * CHUNK * DIMS);
    u32x4 g0;
    g0[0] = 1u;                                            // count=1, is_restore=0, gather off
    g0[1] = (unsigned)(size_t)(void*)dst;                  // lds_addr (bytes)
    g0[2] = (unsigned)(ga & 0xFFFFFFFFull);                // global_addr[31:0]
    g0[3] = (unsigned)((ga >> 32) & 0x01FFFFFFull)         // global_addr[56:32]
          | (2u << 30);                                    // type=2 ("image")
    i32x8 g1;
    g1[0] = (2 << 16);                                     // wg_mask=0, data_size=2 (4B), no flags
    g1[1] = (DIMS  << 16);                                 // tensor_dim0[15:0]=48
    g1[2] = (CHUNK << 16);                                 // tensor_dim0[31:16]=0, tensor_dim1[15:0]=64
    g1[3] = (DIMS  << 16);                                 // tensor_dim1[31:16]=0, tile_dim0=48
    g1[4] = CHUNK;                                         // tile_dim1=64, tile_dim2=0 (2D)
    g1[5] = DIMS;                                          // tensor_dim0_stride[31:0]=48
    g1[6] = 0;                                             // stride0 hi, tensor_dim1_stride lo
    g1[7] = 0;                                             // tensor_dim1_stride hi
    i32x4 z4 = {0, 0, 0, 0};                               // groups 2/3: unused for 2D tile
    i32x8 z8 = {0, 0, 0, 0, 0, 0, 0, 0};                   // trailing group: unused for 2D tile
    __builtin_amdgcn_tensor_load_to_lds(g0, g1, z4, z4, z8, 0);
}
#endif

// ---------------------------------------------------------------------------
// Path B (fallback): async global->LDS b128 copies (ASYNCcnt-tracked).
// ---------------------------------------------------------------------------
__device__ __forceinline__ void async_copy_b128(unsigned lds_byte_addr, const void* gsrc) {
    asm volatile("global_load_async_to_lds_b128 %0, %1, off"
                 :: "v"(lds_byte_addr), "v"(gsrc)
                 : "memory");
}

__device__ __forceinline__ void async_prefetch_chunk(const float* row, int c, float* dst, int tid) {
    const char* src      = (const char*)(row) + (size_t)c * CHUNK_BYTES;
    unsigned    lds_base = (unsigned)(size_t)(void*)dst;
#pragma unroll
    for (int k = 0; k < CHUNK_BYTES / (BLOCK * 16); ++k) {
        int off = (tid + k * BLOCK) * 16;
        async_copy_b128(lds_base + (unsigned)off, src + off);
    }
}

// Unified issue + wait
__device__ __forceinline__ void load_chunk(const float* row, int c, float* dst, int tid) {
#if USE_TDM
    if (tid == 0) tdm_load_chunk(row, c, dst);   // wave 0 only; TDM ignores EXEC
#else
    async_prefetch_chunk(row, c, dst, tid);      // both waves share the copy
#endif
}

__device__ __forceinline__ void wait_dma0() {
#if USE_TDM
    __builtin_amdgcn_s_wait_tensorcnt(0);        // wave1: count already 0, passes through
#else
#if defined(__has_builtin)
#if __has_builtin(__builtin_amdgcn_s_wait_asynccnt)
    __builtin_amdgcn_s_wait_asynccnt(0);
    return;
#endif
#endif
    asm volatile("s_wait_asynccnt 0x0" ::: "memory");
#endif
}

__global__ __launch_bounds__(BLOCK)
void crf_viterbi_kernel(const float* __restrict__ logits,
                        const int*   __restrict__ lens,
                        const float* __restrict__ trans,
                        float*       __restrict__ out)
{
    __shared__ __align__(16) float  s_emit[2][CHUNK * DIMS];   // 24576 B, double-buffered
    __shared__ __align__(16) float  s_alpha[2][DIMS];          // double-buffered alpha
    __shared__ unsigned char        s_bp[(TMAX - 1) * DIMS];   // 24528 B backpointers (LDS only)
    __shared__ int                  s_pred[TMAX];

    const int b   = blockIdx.x;
    const int tid = threadIdx.x;
    const int j   = tid;                 // state owned by this thread (j < DIMS active)
    const int len = lens[b];
    const float* row = logits + (size_t)b * TMAX * DIMS;

    // --- one-time: thread j caches trans[:, j] in registers (coalesced loads) ---
    const int jj = (j < DIMS) ? j : (DIMS - 1);
    float tc[DIMS];
#pragma unroll
    for (int i = 0; i < DIMS; ++i) tc[i] = trans[i * DIMS + jj];

    const int nchunk = (len + CHUNK - 1) / CHUNK;   // only fetch what the scan needs

    load_chunk(row, 0, &s_emit[0][0], tid);         // kick off chunk 0

    int cur = 0;
    for (int c = 0; c < nchunk; ++c) {
        wait_dma0();            // issuing wave's DMA complete
        __syncthreads();        // publish staged chunk to the other wave

        if (c + 1 < nchunk)     // overlap next chunk with this chunk's recurrence
            load_chunk(row, c + 1, &s_emit[(c + 1) & 1][0], tid);

        const float* eb = &s_emit[c & 1][0];
        int t0 = c * CHUNK;
        int t1 = (len < t0 + CHUNK) ? len : (t0 + CHUNK);

        if (c == 0) {
            if (j < DIMS) s_alpha[0][j] = eb[j];    // alpha0 = logits[b,0,:]
            __syncthreads();
            t0 = 1;
        }

        for (int tt = t0; tt < t1; ++tt) {
            if (j < DIMS) {
                const float4* ac4 = reinterpret_cast<const float4*>(s_alpha[cur]);
                float best = -INFINITY; int arg = 0;
#pragma unroll
                for (int i4 = 0; i4 < DIMS / 4; ++i4) {
                    float4 a = ac4[i4];                        // LDS broadcast (uniform addr)
                    float s0 = a.x + tc[4 * i4 + 0];
                    float s1 = a.y + tc[4 * i4 + 1];
                    float s2 = a.z + tc[4 * i4 + 2];
                    float s3 = a.w + tc[4 * i4 + 3];
                    if (s0 > best) { best = s0; arg = 4 * i4 + 0; }   // first-max == jnp.argmax
                    if (s1 > best) { best = s1; arg = 4 * i4 + 1; }
                    if (s2 > best) { best = s2; arg = 4 * i4 + 2; }
                    if (s3 > best) { best = s3; arg = 4 * i4 + 3; }
                }
                float na = best + eb[(tt - c * CHUNK) * DIMS + j];
                s_alpha[cur ^ 1][j]        = na;
                s_bp[(tt - 1) * DIMS + j]  = (unsigned char)arg;
            }
            __syncthreads();
            cur ^= 1;
        }
    }

    wait_dma0();
    __syncthreads();

    // --- serial backtrack entirely in LDS ---
    if (tid == 0) {
        const float* af = s_alpha[cur];
        float best = af[0]; int arg = 0;
        for (int i = 1; i < DIMS; ++i)
            if (af[i] > best) { best = af[i]; arg = i; }
        int tag = arg;
        for (int tt = len - 1; tt >= 1; --tt) {
            s_pred[tt] = tag;
            tag = (int)s_bp[(tt - 1) * DIMS + tag];
        }
        s_pred[0] = tag;
    }
    __syncthreads();

    // --- coalesced output; padded positions -> 0 (reference mask) ---
    for (int tt = tid; tt < TMAX; tt += BLOCK)
        out[(size_t)b * TMAX + tt] = (tt < len) ? (float)s_pred[tt] : 0.0f;
}

extern "C" void kernel_launch(void* const* d_in, const int* in_sizes, int n_in,
                              void* d_out, int out_size, void* d_ws, size_t ws_size,
                              hipStream_t stream) {
    const float* logits = (const float*)d_in[0];   // [B, T, DIM] f32
    const int*   lens   = (const int*)  d_in[1];   // [B] i32
    const float* trans  = (const float*)d_in[2];   // [DIM, DIM] f32
    float*       out    = (float*)d_out;           // [B, T] (int tags stored as f32)

    const int B = in_sizes[1];                     // 512
    crf_viterbi_kernel<<<dim3(B), dim3(BLOCK), 0, stream>>>(logits, lens, trans, out);
}